// PiecewiseLinearEmbedding_9792525434945
// MI455X (gfx1250) — compile-verified
//
#include <hip/hip_runtime.h>

// Piecewise-linear embedding:
//   out[t, :] = cumsum(W, axis=1)[:, x[t]] + b      (t = flattened 8192*200 tokens)
// = gather from a fused 129x64 lookup table.  Pure store-bandwidth-bound:
// 419 MB out / 23.3 TB/s ~= 18 us floor.  Strategy: table in LDS (async
// global->LDS, ASYNCcnt), ds_load_b128 gather, nontemporal b128 stores.

#define NUM_BINS 129                  // NUM_FEATURE + 1
#define VDIM     64
#define TBL_ELEMS (NUM_BINS * VDIM)   // 8256 floats = 33,024 bytes

// native clang vectors (HIP's float4 is a class; builtins need real vectors)
typedef float f4  __attribute__((ext_vector_type(4)));
typedef int   v4i __attribute__((ext_vector_type(4)));

// ---------------------------------------------------------------------------
// Kernel A: build fused table  T[j*64 + d] = sum_{k<=j} W[d,k] + b[d]
// W is (64, 129) row-major.  One block, 64 threads, serial cumsum per d.
// ---------------------------------------------------------------------------
__global__ void ple_build_table(const float* __restrict__ W,
                                const float* __restrict__ bias,
                                float* __restrict__ table_g) {
  const int d = threadIdx.x;
  if (d < VDIM) {
    float acc = bias[d];
#pragma unroll 1
    for (int j = 0; j < NUM_BINS; ++j) {
      acc += W[d * NUM_BINS + j];
      table_g[j * VDIM + d] = acc;   // coalesced across d per j
    }
  }
}

// ---------------------------------------------------------------------------
// Kernel B: gather.  16 lanes cooperate on one token: lane l writes a
// float4 at out[t*64 + l*4] from LDS table row x[t].  256 B per token,
// fully coalesced; LDS row is 64-bank aligned -> conflict-free b128 reads.
// ---------------------------------------------------------------------------
__global__ __launch_bounds__(256) void ple_gather(const int* __restrict__ x,
                                                  const float* __restrict__ table_g,
                                                  float* __restrict__ out,
                                                  int ntok) {
  __shared__ __align__(16) float table[TBL_ELEMS];

  // ---- stage table into LDS (CDNA5 async global->LDS path) -----------------
#if defined(__AMDGCN__) && __has_builtin(__builtin_amdgcn_global_load_async_to_lds_b128)
  {
    const int nchunks = TBL_ELEMS / 4;          // 2064 x 16B
    for (int i = threadIdx.x; i < (int)nchunks; i += blockDim.x) {
      __builtin_amdgcn_global_load_async_to_lds_b128(
          (__attribute__((address_space(1))) v4i*)(table_g + i * 4),
          (__attribute__((address_space(3))) v4i*)(table + i * 4),
          /*offset=*/0, /*cpol=*/0);
    }
#if __has_builtin(__builtin_amdgcn_s_wait_asynccnt)
    __builtin_amdgcn_s_wait_asynccnt(0);
#else
    asm volatile("s_wait_asynccnt 0" ::: "memory");
#endif
  }
#else
  {
    const int nchunks = TBL_ELEMS / 4;
    for (int i = threadIdx.x; i < (int)nchunks; i += blockDim.x) {
      ((f4*)table)[i] = ((const f4*)table_g)[i];
    }
  }
#endif
  __syncthreads();

  // ---- gather + stream out --------------------------------------------------
  const int lane   = threadIdx.x & 15;                               // 0..15
  const int group  = (int)((blockIdx.x * blockDim.x + threadIdx.x) >> 4);
  const int stride = (int)((gridDim.x * blockDim.x) >> 4);

  for (int t = group; t < ntok; t += stride) {
    const int xv = x[t];                                             // broadcast load
    const f4 v = *(const f4*)(table + xv * VDIM + lane * 4);         // ds_load_b128
    // 419 MB > 192 MB L2: write-once stream, use nontemporal hint
    __builtin_nontemporal_store(v, (f4*)(out + (size_t)t * VDIM + lane * 4));
  }
}

// ---------------------------------------------------------------------------
extern "C" void kernel_launch(void* const* d_in, const int* in_sizes, int n_in,
                              void* d_out, int out_size, void* d_ws, size_t ws_size,
                              hipStream_t stream) {
  const int*   x = (const int*)  d_in[0];   // (8192, 200) int32
  const float* W = (const float*)d_in[1];   // (64, 129) f32
  const float* b = (const float*)d_in[2];   // (64,) f32
  float* out     = (float*)d_out;           // (8192, 200, 64) f32
  float* table_g = (float*)d_ws;            // 33 KB scratch

  const int ntok = in_sizes[0];             // 8192*200 tokens (x is 1 int/token)

  ple_build_table<<<1, 64, 0, stream>>>(W, b, table_g);

  int blocks = (ntok + 15) / 16;            // 16 tokens per block per iteration
  if (blocks > 4096) blocks = 4096;         // grid-stride; amortize LDS table fill
  ple_gather<<<blocks, 256, 0, stream>>>(x, table_g, out, ntok);
}